// GraphConvolution_28578712388014
// MI455X (gfx1250) — compile-verified
//
#include <hip/hip_runtime.h>
#include <math.h>

#define NNODES 50000
#define NEDGES 800000
#define DIM    128
#define NEG_SLOPE 0.2f
#define L2_EPS 1e-12f
#define LDS_STRIDE 132   // 128 + 4 pad: removes 16-way bank conflicts on A-fragment reads

typedef float v2f __attribute__((ext_vector_type(2)));
typedef float v8f __attribute__((ext_vector_type(8)));

// ---------------------------------------------------------------------------
// Kernel 1: edge scatter. One wave32 per edge; lane L handles columns
// [4L, 4L+4) via a float4 load of the source row and 4 native f32 atomics
// into the destination accumulator. Lane 0 bumps the degree counter.
// ---------------------------------------------------------------------------
__global__ __launch_bounds__(256)
void gc_edge_scatter(const float* __restrict__ node_fts,
                     const int*   __restrict__ edges,
                     float*       __restrict__ seg_sum,
                     float*       __restrict__ cnt) {
    unsigned gid  = blockIdx.x * blockDim.x + threadIdx.x;
    unsigned e    = gid >> 5;
    unsigned lane = gid & 31u;
    if (e >= NEDGES) return;

    int dst = edges[e];            // edges[0][e] : aggregation target
    int src = edges[NEDGES + e];   // edges[1][e] : neighbor feature index

    const float4 v = *(const float4*)(node_fts + (size_t)src * DIM + lane * 4);
    float* o = seg_sum + (size_t)dst * DIM + lane * 4;
    unsafeAtomicAdd(o + 0, v.x);
    unsafeAtomicAdd(o + 1, v.y);
    unsafeAtomicAdd(o + 2, v.z);
    unsafeAtomicAdd(o + 3, v.w);
    if (lane == 0) unsafeAtomicAdd(cnt + dst, 1.0f);
}

// ---------------------------------------------------------------------------
// Kernel 2: fused dual-GEMM + LeakyReLU + LeakyReLU + row L2-normalize.
// Block = 256 threads = 8 waves, handling a 16-node tile.
// Wave w computes output columns [16w, 16w+16) of BOTH GEMMs with
// V_WMMA_F32_16X16X4_F32, K looped 32x over D=128.
//   out[m][n] = sum_k X[m][k] * W[n][k]   (A = X tile, B = W^T)
// A-fragment (16x4 f32): lane holds row m=lane%16, K-pair kk=2*(lane/16)
// B-fragment (4x16 f32): lane holds col n=lane%16, same K-pair
//  -> both are contiguous float2 loads at (row*stride + t*4 + kk).
// C/D: VGPR r -> row r + 8*(lane/16), col = lane%16.
// ---------------------------------------------------------------------------
__global__ __launch_bounds__(256)
void gc_node_wmma(const float* __restrict__ node_fts,
                  const float* __restrict__ W_self,
                  const float* __restrict__ W_neigh,
                  const float* __restrict__ seg_sum,
                  const float* __restrict__ cnt,
                  float*       __restrict__ out) {
    __shared__ float xs[16 * LDS_STRIDE];   // self features tile
    __shared__ float xn[16 * LDS_STRIDE];   // neighbor-mean tile
    __shared__ float ob[16 * LDS_STRIDE];   // fused output tile
    __shared__ float scale[16];

    const unsigned tid      = threadIdx.x;
    const unsigned wave     = tid >> 5;         // 0..7 -> column tile
    const unsigned lane     = tid & 31u;
    const unsigned nodeBase = blockIdx.x * 16;  // NNODES % 16 == 0

    // ---- stage tiles into LDS: 2048 floats each, 8 per thread -------------
    {
        unsigned idx  = tid * 8;
        unsigned row  = idx >> 7;       // / 128
        unsigned col  = idx & 127u;
        unsigned node = nodeBase + row;
        float inv = 1.0f / fmaxf(cnt[node], 1.0f);

        const float4* ss = (const float4*)(node_fts + (size_t)node * DIM + col);
        const float4* sn = (const float4*)(seg_sum  + (size_t)node * DIM + col);
        float4 a0 = ss[0], a1 = ss[1];
        float4 b0 = sn[0], b1 = sn[1];

        float* ps = xs + row * LDS_STRIDE + col;
        float* pn = xn + row * LDS_STRIDE + col;
        ps[0] = a0.x; ps[1] = a0.y; ps[2] = a0.z; ps[3] = a0.w;
        ps[4] = a1.x; ps[5] = a1.y; ps[6] = a1.z; ps[7] = a1.w;
        pn[0] = b0.x * inv; pn[1] = b0.y * inv; pn[2] = b0.z * inv; pn[3] = b0.w * inv;
        pn[4] = b1.x * inv; pn[5] = b1.y * inv; pn[6] = b1.z * inv; pn[7] = b1.w * inv;
    }
    __syncthreads();

    const unsigned m    = lane & 15u;          // A row within tile
    const unsigned kk   = (lane >> 4) << 1;    // 0 or 2: K sub-pair
    const unsigned ncol = wave * 16 + m;       // global output column (W row)

    const float* wsRow = W_self  + (size_t)ncol * DIM;
    const float* wnRow = W_neigh + (size_t)ncol * DIM;
    const float* aS    = xs + m * LDS_STRIDE;
    const float* aN    = xn + m * LDS_STRIDE;

    v8f accS = {};
    v8f accN = {};

#pragma unroll 4
    for (int t = 0; t < 32; ++t) {
        const unsigned k0 = t * 4 + kk;
        v2f fa_s = *(const v2f*)(aS + k0);
        v2f fb_s = *(const v2f*)(wsRow + k0);
        v2f fa_n = *(const v2f*)(aN + k0);
        v2f fb_n = *(const v2f*)(wnRow + k0);
        accS = __builtin_amdgcn_wmma_f32_16x16x4_f32(
                   false, fa_s, false, fb_s, (short)0, accS, false, false);
        accN = __builtin_amdgcn_wmma_f32_16x16x4_f32(
                   false, fa_n, false, fb_n, (short)0, accN, false, false);
    }

    // ---- epilogue: leaky(neigh), sum with self, leaky, stage to LDS -------
    const unsigned rbase = (lane >> 4) << 3;   // 0 or 8
#pragma unroll
    for (int r = 0; r < 8; ++r) {
        float nf = accN[r];
        nf = nf > 0.0f ? nf : NEG_SLOPE * nf;          // leaky on neighbor GEMM
        float u = accS[r] + nf;
        u = u > 0.0f ? u : NEG_SLOPE * u;              // leaky on update
        ob[(rbase + r) * LDS_STRIDE + ncol] = u;
    }
    __syncthreads();

    // ---- per-row L2 norm across all 128 columns ---------------------------
    if (tid < 16) {
        const float* pr = ob + tid * LDS_STRIDE;
        float s = 0.0f;
#pragma unroll 8
        for (int c = 0; c < DIM; ++c) { float v = pr[c]; s += v * v; }
        scale[tid] = 1.0f / fmaxf(sqrtf(s), L2_EPS);
    }
    __syncthreads();

    // ---- scaled store: 8 floats per thread --------------------------------
    {
        unsigned idx = tid * 8;
        unsigned row = idx >> 7;
        unsigned col = idx & 127u;
        float sc = scale[row];
        const float* pb = ob + row * LDS_STRIDE + col;
        float4 o0 = make_float4(pb[0] * sc, pb[1] * sc, pb[2] * sc, pb[3] * sc);
        float4 o1 = make_float4(pb[4] * sc, pb[5] * sc, pb[6] * sc, pb[7] * sc);
        float4* po = (float4*)(out + (size_t)(nodeBase + row) * DIM + col);
        po[0] = o0;
        po[1] = o1;
    }
}

// ---------------------------------------------------------------------------
extern "C" void kernel_launch(void* const* d_in, const int* in_sizes, int n_in,
                              void* d_out, int out_size, void* d_ws, size_t ws_size,
                              hipStream_t stream) {
    const float* node_fts = (const float*)d_in[0];
    const int*   edges    = (const int*)  d_in[1];
    // d_in[2] = edge_fts : unused by the reference computation
    const float* W_self   = (const float*)d_in[3];
    const float* W_neigh  = (const float*)d_in[4];
    float*       out      = (float*)d_out;

    float* seg_sum = (float*)d_ws;                         // [NNODES * DIM]
    float* cnt     = seg_sum + (size_t)NNODES * DIM;       // [NNODES]

    // zero the accumulators every launch (harness poisons d_ws once)
    hipMemsetAsync(d_ws, 0, ((size_t)NNODES * DIM + NNODES) * sizeof(float), stream);

    // edge pass: one wave32 per edge
    {
        const unsigned total  = NEDGES * 32u;
        const unsigned block  = 256;
        const unsigned grid   = (total + block - 1) / block;
        gc_edge_scatter<<<grid, block, 0, stream>>>(node_fts, edges, seg_sum, cnt);
    }

    // node pass: 16 nodes per block (NNODES % 16 == 0 -> 3125 blocks)
    {
        const unsigned grid = NNODES / 16;
        gc_node_wmma<<<grid, 256, 0, stream>>>(node_fts, W_self, W_neigh,
                                               seg_sum, cnt, out);
    }
}